// MVmodel_69879117906023
// MI455X (gfx1250) — compile-verified
//
#include <hip/hip_runtime.h>
#include <math.h>

// CDNA5 f32 WMMA: D(16x16,f32) = A(16x4,f32) x B(4x16,f32) + C
typedef __attribute__((ext_vector_type(2))) float v2f;
typedef __attribute__((ext_vector_type(8))) float v8f;

#define WMMA_F32(a, b, c) \
  __builtin_amdgcn_wmma_f32_16x16x4_f32(false, (a), false, (b), (short)0, (c), false, false)

// ---------------- small elementwise kernels ----------------

__global__ __launch_bounds__(256) void k_zero(float* __restrict__ p, int n) {
  int i = blockIdx.x * blockDim.x + threadIdx.x;
  if (i < n) p[i] = 0.0f;
}

__global__ __launch_bounds__(256) void k_degree(const int* __restrict__ dst, int E,
                                                float* __restrict__ deg) {
  int i = blockIdx.x * blockDim.x + threadIdx.x;
  if (i < E) atomicAdd(&deg[dst[i]], 1.0f);
}

// dinv[i] = rsqrt(indeg[i] + 1)   (self-loop included; always > 0)
__global__ __launch_bounds__(256) void k_dinv(float* __restrict__ deg, int n) {
  int i = blockIdx.x * blockDim.x + threadIdx.x;
  if (i < n) deg[i] = rsqrtf(deg[i] + 1.0f);
}

// x1 = prelu(dinv[row]*agg + b[col]);  H = 1<<hshift
__global__ __launch_bounds__(256) void k_finalize_prelu(
    const float* __restrict__ agg, const float* __restrict__ dinv,
    const float* __restrict__ b, const float* __restrict__ a_ptr,
    float* __restrict__ out, long total, int hshift) {
  long i = (long)blockIdx.x * blockDim.x + threadIdx.x;
  if (i >= total) return;
  int row = (int)(i >> hshift);
  int col = (int)(i & ((1 << hshift) - 1));
  float v = dinv[row] * agg[i] + b[col];
  float a = a_ptr[0];
  out[i] = (v >= 0.0f) ? v : a * v;
}

// ---------------- edge scatter: agg[dst] += hs[src] ----------------
// one wave (32 lanes) per edge; strided over H features -> coalesced b32
// loads + coalesced global_atomic_add_f32.
__global__ __launch_bounds__(256) void k_scatter(
    const int* __restrict__ src, const int* __restrict__ dst, int E,
    const float* __restrict__ hs, float* __restrict__ agg, int H) {
  int e = blockIdx.x * (blockDim.x >> 5) + (threadIdx.x >> 5);
  if (e >= E) return;
  int lane = threadIdx.x & 31;
  const float* hp = hs + (size_t)src[e] * H;
  float* ap = agg + (size_t)dst[e] * H;
  for (int c = lane; c < H; c += 32) atomicAdd(&ap[c], hp[c]);
}

// ---------------- WMMA GEMM + dinv-scale epilogue ----------------
// hs = agg = dinv[row] * (X @ W).  X:[N,K] row-major, W:[K,H] row-major.
// Block: (32, H/16) waves; wave y owns cols [16y,16y+16); block owns 16 rows.
// Requires N%16==0, K%4==0, H%16==0, all-lanes-active (WMMA EXEC rule).
__global__ __launch_bounds__(256) void k_gemm_scale(
    const float* __restrict__ X, const float* __restrict__ W,
    const float* __restrict__ dinv, float* __restrict__ hs,
    float* __restrict__ agg, int K, int H) {
  extern __shared__ float ldsX[];           // 16 rows x (K+4) pad stride
  const int KP = K + 4;                     // pad breaks stride-K bank conflicts
  const int row0 = blockIdx.x << 4;
  const int tid = threadIdx.y * 32 + threadIdx.x;
  const int T = blockDim.y * 32;
  const int k4 = K >> 2;
  for (int i = tid; i < 16 * k4; i += T) {  // stage 16xK tile of X into LDS
    int r = i / k4, c = i - r * k4;
    ((float4*)(ldsX + r * KP))[c] = ((const float4*)(X + (size_t)(row0 + r) * K))[c];
  }
  __syncthreads();

  const int lane = threadIdx.x;
  const int half = lane >> 4;               // 0: lanes 0-15, 1: lanes 16-31
  const int lm = lane & 15;
  const int n0 = threadIdx.y << 4;
  v8f acc = {};
  for (int k = 0; k < K; k += 4) {
    const int ka = k + 2 * half;            // A/B frag K-split per half-wave
    v2f a, b;
    a.x = ldsX[lm * KP + ka];
    a.y = ldsX[lm * KP + ka + 1];
    b.x = W[(size_t)ka * H + n0 + lm];
    b.y = W[(size_t)(ka + 1) * H + n0 + lm];
    acc = WMMA_F32(a, b, acc);
  }
#pragma unroll
  for (int r = 0; r < 8; ++r) {             // D: VGPR r -> row r + 8*half
    int row = row0 + r + 8 * half;
    float v = acc[r] * dinv[row];
    size_t off = (size_t)row * H + n0 + lm;
    hs[off] = v;                            // message value
    agg[off] = v;                           // self-loop init (saves a copy pass)
  }
}

// ---------------- fused projection head: elu(X@P1+pb1)@P2+pb2 ----------------
// X:[N,64]. Block: (32,4) waves, 16 rows per block. Both GEMMs through LDS.
__global__ __launch_bounds__(128) void k_project(
    const float* __restrict__ X, const float* __restrict__ P1,
    const float* __restrict__ pb1, const float* __restrict__ P2,
    const float* __restrict__ pb2, float* __restrict__ out) {
  __shared__ float ldsX[16 * 68];
  __shared__ float ldsT[16 * 68];
  const int row0 = blockIdx.x << 4;
  const int tid = threadIdx.y * 32 + threadIdx.x;
  for (int i = tid; i < 16 * 16; i += 128) {  // 16x64 floats as float4
    int r = i >> 4, c = i & 15;
    ((float4*)(ldsX + r * 68))[c] = ((const float4*)(X + (size_t)(row0 + r) * 64))[c];
  }
  __syncthreads();

  const int lane = threadIdx.x;
  const int half = lane >> 4;
  const int lm = lane & 15;
  const int n0 = threadIdx.y << 4;

  v8f acc = {};
  for (int k = 0; k < 64; k += 4) {
    const int ka = k + 2 * half;
    v2f a, b;
    a.x = ldsX[lm * 68 + ka];
    a.y = ldsX[lm * 68 + ka + 1];
    b.x = P1[ka * 64 + n0 + lm];
    b.y = P1[(ka + 1) * 64 + n0 + lm];
    acc = WMMA_F32(a, b, acc);
  }
  float bias1 = pb1[n0 + lm];
#pragma unroll
  for (int r = 0; r < 8; ++r) {             // ELU, stash in LDS for 2nd GEMM
    float v = acc[r] + bias1;
    v = (v > 0.0f) ? v : expm1f(v);
    ldsT[(r + 8 * half) * 68 + n0 + lm] = v;
  }
  __syncthreads();

  v8f acc2 = {};
  for (int k = 0; k < 64; k += 4) {
    const int ka = k + 2 * half;
    v2f a, b;
    a.x = ldsT[lm * 68 + ka];
    a.y = ldsT[lm * 68 + ka + 1];
    b.x = P2[ka * 64 + n0 + lm];
    b.y = P2[(ka + 1) * 64 + n0 + lm];
    acc2 = WMMA_F32(a, b, acc2);
  }
  float bias2 = pb2[n0 + lm];
#pragma unroll
  for (int r = 0; r < 8; ++r)
    out[(size_t)(row0 + r + 8 * half) * 64 + n0 + lm] = acc2[r] + bias2;
}

// ---------------- host-side launch ----------------

extern "C" void kernel_launch(void* const* d_in, const int* in_sizes, int n_in,
                              void* d_out, int out_size, void* d_ws, size_t ws_size,
                              hipStream_t stream) {
  const float* x   = (const float*)d_in[0];   // [N,128]
  const int*   ei  = (const int*)d_in[1];     // [2,E]
  const float* W1  = (const float*)d_in[2];   // [128,128]
  const float* b1  = (const float*)d_in[3];   // [128]
  const float* W2  = (const float*)d_in[4];   // [128,64]
  const float* b2  = (const float*)d_in[5];   // [64]
  const float* pa  = (const float*)d_in[6];   // scalar
  const float* P1  = (const float*)d_in[7];   // [64,64]
  const float* pb1 = (const float*)d_in[8];   // [64]
  const float* P2  = (const float*)d_in[9];   // [64,64]
  const float* pb2 = (const float*)d_in[10];  // [64]

  const int IN = 128, HID = 128, OUT = 64;
  const int N = in_sizes[0] / IN;             // 50000 (multiple of 16)
  const int E = in_sizes[1] / 2;              // 800000
  const int* srcI = ei;
  const int* dstI = ei + E;

  // workspace layout (floats), 16B-aligned chunks
  float* ws = (float*)d_ws;
  const long Np = (N + 3) & ~3L;
  float* dinv = ws;                 // N
  float* hs   = dinv + Np;          // N*128
  float* agg  = hs + (long)Np * HID;// N*128
  float* x1   = agg + (long)Np * HID;// N*128
  float* x2   = x1 + (long)Np * HID; // N*64

  const int TB = 256;
  // degrees -> dinv
  k_zero<<<(N + TB - 1) / TB, TB, 0, stream>>>(dinv, N);
  k_degree<<<(E + TB - 1) / TB, TB, 0, stream>>>(dstI, E, dinv);
  k_dinv<<<(N + TB - 1) / TB, TB, 0, stream>>>(dinv, N);

  const int rowBlocks = N / 16;
  const int edgeBlocks = (E + 7) / 8;  // 8 waves/block, 1 edge/wave
  const size_t shmem128 = 16 * (128 + 4) * sizeof(float);

  // layer 1: hs = agg = dinv*(x@W1); scatter; x1 = prelu(dinv*agg + b1)
  k_gemm_scale<<<rowBlocks, dim3(32, 8), shmem128, stream>>>(x, W1, dinv, hs, agg, IN, HID);
  k_scatter<<<edgeBlocks, TB, 0, stream>>>(srcI, dstI, E, hs, agg, HID);
  k_finalize_prelu<<<(int)(((long)N * HID + TB - 1) / TB), TB, 0, stream>>>(
      agg, dinv, b1, pa, x1, (long)N * HID, 7);

  // layer 2 (reuse hs/agg): hs = agg = dinv*(x1@W2); scatter; x2 = prelu(...)
  k_gemm_scale<<<rowBlocks, dim3(32, 4), shmem128, stream>>>(x1, W2, dinv, hs, agg, HID, OUT);
  k_scatter<<<edgeBlocks, TB, 0, stream>>>(srcI, dstI, E, hs, agg, OUT);
  k_finalize_prelu<<<(int)(((long)N * OUT + TB - 1) / TB), TB, 0, stream>>>(
      agg, dinv, b2, pa, x2, (long)N * OUT, 6);

  // projection head -> d_out
  k_project<<<rowBlocks, dim3(32, 4), 0, stream>>>(x2, P1, pb1, P2, pb2, (float*)d_out);
}